// Sage_36344013259382
// MI455X (gfx1250) — compile-verified
//
#include <hip/hip_runtime.h>
#include <hip/hip_bf16.h>

// Problem constants (match reference)
#define N0    262144
#define N1_   65536
#define N2_   16384
#define E1_   1048576
#define E2_   262144
#define IN_C  128
#define HID_C 256
#define OUT_C 64

typedef __attribute__((ext_vector_type(2))) float v2f;
typedef __attribute__((ext_vector_type(8))) float v8f;

// ---------------------------------------------------------------------------
// Degree histogram: 1 int atomic per edge.
// ---------------------------------------------------------------------------
__global__ void sage_count_deg(const int* __restrict__ dst, int* __restrict__ deg,
                               int E) {
    int e = blockIdx.x * blockDim.x + threadIdx.x;
    if (e < E) atomicAdd(&deg[dst[e]], 1);
}

// ---------------------------------------------------------------------------
// Single-block exclusive scan (Hillis-Steele in LDS, carry across chunks).
// ---------------------------------------------------------------------------
__global__ void sage_excl_scan(const int* __restrict__ in, int* __restrict__ out,
                               int n) {
    __shared__ int buf[1024];
    __shared__ int carry_s;
    const int tid = threadIdx.x;
    if (tid == 0) carry_s = 0;
    __syncthreads();
    for (int base = 0; base < n; base += 1024) {
        int i = base + tid;
        int v = (i < n) ? in[i] : 0;
        buf[tid] = v;
        __syncthreads();
        for (int off = 1; off < 1024; off <<= 1) {
            int t = (tid >= off) ? buf[tid - off] : 0;
            __syncthreads();
            buf[tid] += t;
            __syncthreads();
        }
        int incl = buf[tid];
        int c = carry_s;
        if (i < n) out[i] = c + incl - v;
        __syncthreads();
        if (tid == 0) carry_s = c + buf[1023];
        __syncthreads();
    }
}

// ---------------------------------------------------------------------------
// Bucket edges by destination: esrc[offs[d] + ticket] = src[e].
// ---------------------------------------------------------------------------
__global__ void sage_bucket_edges(const int* __restrict__ src,
                                  const int* __restrict__ dst,
                                  const int* __restrict__ offs,
                                  int* __restrict__ fill,
                                  int* __restrict__ esrc, int E) {
    int e = blockIdx.x * blockDim.x + threadIdx.x;
    if (e >= E) return;
    int d = dst[e];
    int pos = offs[d] + atomicAdd(&fill[d], 1);
    esrc[pos] = src[e];
}

// ---------------------------------------------------------------------------
// Gather-reduce mean: one wave32 per destination row. Lanes batch-load 32
// edge ids (coalesced), __shfl-broadcast each, accumulate the source row in
// registers (32 lanes x float4 x V). Mean folded into the epilogue.
// ---------------------------------------------------------------------------
template <int C>
__global__ void sage_gather_mean(const float* __restrict__ x,
                                 const int* __restrict__ esrc,
                                 const int* __restrict__ offs,
                                 const int* __restrict__ deg,
                                 float* __restrict__ outm, int nDst) {
    const int wave = blockIdx.x * (blockDim.x >> 5) + (threadIdx.x >> 5);
    const int lane = threadIdx.x & 31;
    if (wave >= nDst) return;          // wave-uniform
    const int beg = offs[wave];
    const int d   = deg[wave];
    constexpr int V = C / 128;         // float4 slots per lane
    float4 acc[V];
#pragma unroll
    for (int v = 0; v < V; ++v) acc[v] = make_float4(0.f, 0.f, 0.f, 0.f);

    for (int base = 0; base < d; base += 32) {
        int n = d - base;
        if (n > 32) n = 32;
        int myid = (lane < n) ? esrc[beg + base + lane] : 0;
        for (int j = 0; j < n; ++j) {
            int s = __shfl(myid, j, 32);
            const float4* row = (const float4*)(x + (size_t)s * C);
#pragma unroll
            for (int v = 0; v < V; ++v) {
                float4 t = row[lane + 32 * v];
                acc[v].x += t.x; acc[v].y += t.y;
                acc[v].z += t.z; acc[v].w += t.w;
            }
        }
    }

    const float scale = (d > 0) ? (1.0f / (float)d) : 0.0f;
    float4* orow = (float4*)(outm + (size_t)wave * C);
#pragma unroll
    for (int v = 0; v < V; ++v) {
        float4 t;
        t.x = acc[v].x * scale; t.y = acc[v].y * scale;
        t.z = acc[v].z * scale; t.w = acc[v].w * scale;
        orow[lane + 32 * v] = t;
    }
}

// ---------------------------------------------------------------------------
// Fused SAGE linear with register blocking:
//   out[m,n] = A1[m,:]@B1[n,:] + A2[m,:]@B2[n,:] + bias[n]   (optional ReLU)
// Each wave computes an MT x NT grid of 16x16 tiles via V_WMMA_F32_16X16X4_F32,
// reusing A fragments across NT tiles and B fragments across MT tiles
// (loads/WMMA = (MT+NT)/(MT*NT)).
// f32 A(16x4) fragment: lanes 0-15 M=lane K{k0,k0+1}; lanes 16-31 K{k0+2,k0+3};
// B(4x16) mirrors with N striped; C/D VGPR r = row r (lanes 0-15) / r+8.
// ---------------------------------------------------------------------------
template <int K, int N, int MT, int NT, bool RELU>
__global__ void sage_gemm_two(const float* __restrict__ A1,
                              const float* __restrict__ B1,
                              const float* __restrict__ A2,
                              const float* __restrict__ B2,
                              const float* __restrict__ bias,
                              float* __restrict__ out, int M) {
    const int lane   = threadIdx.x & 31;
    const int waveId = blockIdx.x * (blockDim.x >> 5) + (threadIdx.x >> 5);
    const int groupsN = N / (16 * NT);
    const int gm = waveId / groupsN;
    const int gn = waveId % groupsN;
    if (gm >= M / (16 * MT)) return;   // wave-uniform guard (EXEC stays all-1)

    const int m0  = gm * 16 * MT;
    const int n0  = gn * 16 * NT;
    const int khi = (lane >> 4) * 2;

    const float* a1p[MT];
    const float* a2p[MT];
#pragma unroll
    for (int mt = 0; mt < MT; ++mt) {
        int r = m0 + mt * 16 + (lane & 15);
        a1p[mt] = A1 + (size_t)r * K + khi;
        a2p[mt] = A2 + (size_t)r * K + khi;
    }
    const float* b1p[NT];
    const float* b2p[NT];
#pragma unroll
    for (int nt = 0; nt < NT; ++nt) {
        int c = n0 + nt * 16 + (lane & 15);
        b1p[nt] = B1 + (size_t)c * K + khi;
        b2p[nt] = B2 + (size_t)c * K + khi;
    }

    v8f acc[MT][NT] = {};

#pragma unroll 2
    for (int k0 = 0; k0 < K; k0 += 4) {
        v2f a[MT], b[NT];
#pragma unroll
        for (int mt = 0; mt < MT; ++mt) a[mt] = *(const v2f*)(a1p[mt] + k0);
#pragma unroll
        for (int nt = 0; nt < NT; ++nt) b[nt] = *(const v2f*)(b1p[nt] + k0);
#pragma unroll
        for (int mt = 0; mt < MT; ++mt)
#pragma unroll
            for (int nt = 0; nt < NT; ++nt)
                acc[mt][nt] = __builtin_amdgcn_wmma_f32_16x16x4_f32(
                    false, a[mt], false, b[nt], (short)0, acc[mt][nt],
                    false, false);
    }
#pragma unroll 2
    for (int k0 = 0; k0 < K; k0 += 4) {
        v2f a[MT], b[NT];
#pragma unroll
        for (int mt = 0; mt < MT; ++mt) a[mt] = *(const v2f*)(a2p[mt] + k0);
#pragma unroll
        for (int nt = 0; nt < NT; ++nt) b[nt] = *(const v2f*)(b2p[nt] + k0);
#pragma unroll
        for (int mt = 0; mt < MT; ++mt)
#pragma unroll
            for (int nt = 0; nt < NT; ++nt)
                acc[mt][nt] = __builtin_amdgcn_wmma_f32_16x16x4_f32(
                    false, a[mt], false, b[nt], (short)0, acc[mt][nt],
                    false, false);
    }

    const int rbase = ((lane >> 4) << 3);        // +8 for upper half-wave
#pragma unroll
    for (int nt = 0; nt < NT; ++nt) {
        const int ocol = n0 + nt * 16 + (lane & 15);
        const float bb = bias[ocol];
#pragma unroll
        for (int mt = 0; mt < MT; ++mt) {
#pragma unroll
            for (int r = 0; r < 8; ++r) {
                float v = acc[mt][nt][r] + bb;
                if (RELU) v = fmaxf(v, 0.0f);
                out[(size_t)(m0 + mt * 16 + rbase + r) * N + ocol] = v;
            }
        }
    }
}

// ---------------------------------------------------------------------------
// In-place log-softmax over 64 columns; one wave32 per row, 2 cols per lane.
// ---------------------------------------------------------------------------
__global__ void sage_logsoftmax64(float* __restrict__ out, int rows) {
    int wave = blockIdx.x * (blockDim.x >> 5) + (threadIdx.x >> 5);
    int lane = threadIdx.x & 31;
    if (wave >= rows) return;
    float* p = out + (size_t)wave * 64;
    float a = p[lane];
    float b = p[lane + 32];
    float m = fmaxf(a, b);
#pragma unroll
    for (int off = 16; off > 0; off >>= 1)
        m = fmaxf(m, __shfl_xor(m, off, 32));
    float s = expf(a - m) + expf(b - m);
#pragma unroll
    for (int off = 16; off > 0; off >>= 1)
        s += __shfl_xor(s, off, 32);
    float lse = m + logf(s);
    p[lane]      = a - lse;
    p[lane + 32] = b - lse;
}

// ---------------------------------------------------------------------------
// Host-side orchestration (graph-capture safe: only async ops on `stream`).
// ---------------------------------------------------------------------------
extern "C" void kernel_launch(void* const* d_in, const int* in_sizes, int n_in,
                              void* d_out, int out_size, void* d_ws, size_t ws_size,
                              hipStream_t stream) {
    const float* x    = (const float*)d_in[0];
    const int*   src1 = (const int*)d_in[1];
    const int*   dst1 = (const int*)d_in[2];
    const int*   src2 = (const int*)d_in[3];
    const int*   dst2 = (const int*)d_in[4];
    const float* W1l  = (const float*)d_in[5];
    const float* W1r  = (const float*)d_in[6];
    const float* b1   = (const float*)d_in[7];
    const float* W2l  = (const float*)d_in[8];
    const float* W2r  = (const float*)d_in[9];
    const float* b2   = (const float*)d_in[10];
    float* out = (float*)d_out;

    // Workspace layout (all element counts multiples of 4 -> 16B aligned):
    // [deg1][fill1][deg2][fill2]  <- zeroed each call
    // [offs1][offs2][esrc1][esrc2][aggr1][aggr2][h]
    int* deg1  = (int*)d_ws;
    int* fill1 = deg1 + N1_;
    int* deg2  = fill1 + N1_;
    int* fill2 = deg2 + N2_;
    int* offs1 = fill2 + N2_;
    int* offs2 = offs1 + N1_;
    int* esrc1 = offs2 + N2_;
    int* esrc2 = esrc1 + E1_;
    float* aggr1 = (float*)(esrc2 + E2_);
    float* aggr2 = aggr1 + (size_t)N1_ * IN_C;
    float* h     = aggr2 + (size_t)N2_ * HID_C;

    size_t zbytes = (size_t)(2 * N1_ + 2 * N2_) * sizeof(int);
    hipMemsetAsync(d_ws, 0, zbytes, stream);

    // ---- Layer 1: CSR build + gather-mean + WMMA GEMM (+bias, ReLU) ----
    sage_count_deg<<<E1_ / 256, 256, 0, stream>>>(dst1, deg1, E1_);
    sage_excl_scan<<<1, 1024, 0, stream>>>(deg1, offs1, N1_);
    sage_bucket_edges<<<E1_ / 256, 256, 0, stream>>>(src1, dst1, offs1, fill1,
                                                     esrc1, E1_);
    sage_gather_mean<IN_C><<<N1_ / 8, 256, 0, stream>>>(x, esrc1, offs1, deg1,
                                                        aggr1, N1_);
    {
        // MT=2, NT=4: waves = (65536/32) * (256/64) = 8192
        int waves = (N1_ / 32) * (HID_C / 64);
        sage_gemm_two<IN_C, HID_C, 2, 4, true>
            <<<waves / 8, 256, 0, stream>>>(aggr1, W1l, x, W1r, b1, h, N1_);
    }

    // ---- Layer 2 ----
    sage_count_deg<<<E2_ / 256, 256, 0, stream>>>(dst2, deg2, E2_);
    sage_excl_scan<<<1, 1024, 0, stream>>>(deg2, offs2, N2_);
    sage_bucket_edges<<<E2_ / 256, 256, 0, stream>>>(src2, dst2, offs2, fill2,
                                                     esrc2, E2_);
    sage_gather_mean<HID_C><<<N2_ / 8, 256, 0, stream>>>(h, esrc2, offs2, deg2,
                                                         aggr2, N2_);
    {
        // MT=2, NT=2: waves = (16384/32) * (64/32) = 1024
        int waves = (N2_ / 32) * (OUT_C / 32);
        sage_gemm_two<HID_C, OUT_C, 2, 2, false>
            <<<waves / 8, 256, 0, stream>>>(aggr2, W2l, h, W2r, b2, out, N2_);
    }

    // ---- log_softmax over 64 cols, in place on d_out ----
    sage_logsoftmax64<<<(N2_ + 7) / 8, 256, 0, stream>>>(out, N2_);
}